// ClusterGIN_14697378087625
// MI455X (gfx1250) — compile-verified
//
#include <hip/hip_runtime.h>
#include <hip/hip_bf16.h>

#define NN 100000
#define NE 1600000
#define HID 64
#define OUTC 8

// Swizzled weight layout: pair p = K/2 (0..31), stride 80 pairs per p-row.
// sws[(p*80 + col)*2 + j] = w[(2p+j)*64 + col]  -> one ds_load_b64 per B frag,
// bank = (32p + 2col) mod 64: half-waves hit disjoint bank halves (conflict-free).
#define PSTRIDE 80

typedef float v2f __attribute__((ext_vector_type(2)));
typedef float v8f __attribute__((ext_vector_type(8)));

// ---------------- bulk copy: h = x ----------------
__global__ void __launch_bounds__(256)
copy_f4_kernel(const float4* __restrict__ in, float4* __restrict__ out, int n4) {
  int t = blockIdx.x * blockDim.x + threadIdx.x;
  if (t < n4) out[t] = in[t];
}

// ------------- scatter: h[dst] += feat[src] -------------
// 16 threads per edge, each handles 4 contiguous channels (coalesced float4).
__global__ void __launch_bounds__(256)
scatter_add_kernel(const float* __restrict__ feat, const int* __restrict__ src,
                   const int* __restrict__ dst, float* __restrict__ out, int n_edges) {
  int t = blockIdx.x * blockDim.x + threadIdx.x;
  int e = t >> 4;
  if (e >= n_edges) return;
  int g = t & 15;
  int s = src[e];
  int d = dst[e];
  float4 v = ((const float4*)(feat + (size_t)s * HID))[g];
  float* o = out + (size_t)d * HID + g * 4;
  atomicAdd(o + 0, v.x);
  atomicAdd(o + 1, v.y);
  atomicAdd(o + 2, v.z);
  atomicAdd(o + 3, v.w);
}

// ------------- fused MLP: out = (relu?)(relu(h@w1+b1)@w2+b2) -------------
// 128 threads = 4 wave32; each wave computes a 16x64 output tile via
// V_WMMA_F32_16X16X4_F32 (exact f32). Weights pre-swizzled into B-fragment
// pair order in LDS (single aligned ds_load_b64 per B operand, conflict-free).
// Hidden 16x64 tile bounced through padded LDS (row stride 68).
__global__ void __launch_bounds__(128)
gin_mlp64_kernel(const float* __restrict__ hin,
                 const float* __restrict__ w1, const float* __restrict__ b1,
                 const float* __restrict__ w2, const float* __restrict__ b2,
                 float* __restrict__ hout, int n_nodes, int apply_relu) {
  __shared__ float sw1s[32 * PSTRIDE * 2];   // 20 KB
  __shared__ float sw2s[32 * PSTRIDE * 2];   // 20 KB
  __shared__ float sh[4][16 * 68];           // 17 KB: per-wave hidden tile

  int tid = threadIdx.x;
  for (int i = tid; i < HID * HID; i += 128) {
    int k   = i >> 6;
    int col = i & 63;
    int p   = k >> 1;
    int j   = k & 1;
    int idx = (p * PSTRIDE + col) * 2 + j;
    sw1s[idx] = w1[i];
    sw2s[idx] = w2[i];
  }
  __syncthreads();

  int wave  = tid >> 5;
  int lane  = tid & 31;
  int lhalf = lane >> 4;   // 0: K pair {4k,4k+1}, 1: K pair {4k+2,4k+3}
  int lmod  = lane & 15;
  int row0  = blockIdx.x * 64 + wave * 16;

  const v2f* w1p = (const v2f*)sw1s;
  const v2f* w2p = (const v2f*)sw2s;

  // ---- GEMM1: [16x64] @ [64x64] ----
  v8f acc[4] = {v8f{}, v8f{}, v8f{}, v8f{}};
  int arow = row0 + lmod;
  if (arow >= n_nodes) arow = n_nodes - 1;      // clamp (stores are guarded)
  const v2f* aptr = (const v2f*)(hin + (size_t)arow * HID + 2 * lhalf);

  for (int k = 0; k < 16; ++k) {                // K = 64 in steps of 4
    v2f a = aptr[2 * k];                        // {K=4k(+2lh), K=4k+1(+2lh)}
    int p = 2 * k + lhalf;
#pragma unroll
    for (int nt = 0; nt < 4; ++nt) {
      int col = nt * 16 + lmod;
      v2f b = w1p[p * PSTRIDE + col];
      acc[nt] = __builtin_amdgcn_wmma_f32_16x16x4_f32(
          false, a, false, b, (short)0, acc[nt], false, false);
    }
  }

  // bias + ReLU, spill 16x64 hidden tile to this wave's LDS region
  float* myh = &sh[wave][0];
#pragma unroll
  for (int nt = 0; nt < 4; ++nt) {
    int col = nt * 16 + lmod;
    float bias = b1[col];
#pragma unroll
    for (int r = 0; r < 8; ++r) {
      int row = r + 8 * lhalf;                  // C/D layout: VGPR r -> M = r (+8)
      float v = acc[nt][r] + bias;
      myh[row * 68 + col] = v > 0.f ? v : 0.f;
    }
  }
  __syncthreads();

  // ---- GEMM2: [16x64] @ [64x64] ----
  v8f acc2[4] = {v8f{}, v8f{}, v8f{}, v8f{}};
  for (int k = 0; k < 16; ++k) {
    int kbase = 4 * k + 2 * lhalf;
    v2f a = *(const v2f*)&myh[lmod * 68 + kbase];  // aligned pair, conflict-free
    int p = 2 * k + lhalf;
#pragma unroll
    for (int nt = 0; nt < 4; ++nt) {
      int col = nt * 16 + lmod;
      v2f b = w2p[p * PSTRIDE + col];
      acc2[nt] = __builtin_amdgcn_wmma_f32_16x16x4_f32(
          false, a, false, b, (short)0, acc2[nt], false, false);
    }
  }

  // bias (+ optional trailing ReLU), guarded store
#pragma unroll
  for (int nt = 0; nt < 4; ++nt) {
    int col = nt * 16 + lmod;
    float bias = b2[col];
#pragma unroll
    for (int r = 0; r < 8; ++r) {
      int row = r + 8 * lhalf;
      int node = row0 + row;
      if (node < n_nodes) {
        float v = acc2[nt][r] + bias;
        if (apply_relu) v = v > 0.f ? v : 0.f;
        hout[(size_t)node * HID + col] = v;
      }
    }
  }
}

// ------------- last layer: 64->8 relu 8->8, then log_softmax -------------
__global__ void __launch_bounds__(256)
gin_last_kernel(const float* __restrict__ hin,
                const float* __restrict__ w1, const float* __restrict__ b1,
                const float* __restrict__ w2, const float* __restrict__ b2,
                float* __restrict__ out, int n_nodes) {
  __shared__ float sw1[HID * OUTC];
  __shared__ float sw2[OUTC * OUTC];
  __shared__ float sb1[OUTC];
  __shared__ float sb2[OUTC];
  int tid = threadIdx.x;
  for (int i = tid; i < HID * OUTC; i += 256) sw1[i] = w1[i];
  if (tid < OUTC * OUTC) sw2[tid] = w2[tid];
  if (tid < OUTC) { sb1[tid] = b1[tid]; sb2[tid] = b2[tid]; }
  __syncthreads();

  int node = blockIdx.x * blockDim.x + tid;
  if (node >= n_nodes) return;

  float h[OUTC];
#pragma unroll
  for (int j = 0; j < OUTC; ++j) h[j] = sb1[j];
  const float* xr = hin + (size_t)node * HID;
  for (int c = 0; c < HID; ++c) {
    float xv = xr[c];
#pragma unroll
    for (int j = 0; j < OUTC; ++j) h[j] += xv * sw1[c * OUTC + j];
  }
#pragma unroll
  for (int j = 0; j < OUTC; ++j) h[j] = h[j] > 0.f ? h[j] : 0.f;

  float o[OUTC];
#pragma unroll
  for (int j = 0; j < OUTC; ++j) {
    float a = sb2[j];
#pragma unroll
    for (int k = 0; k < OUTC; ++k) a += h[k] * sw2[k * OUTC + j];
    o[j] = a;
  }

  float m = o[0];
#pragma unroll
  for (int j = 1; j < OUTC; ++j) m = fmaxf(m, o[j]);
  float se = 0.f;
#pragma unroll
  for (int j = 0; j < OUTC; ++j) se += __expf(o[j] - m);
  float lse = __logf(se) + m;
#pragma unroll
  for (int j = 0; j < OUTC; ++j) out[(size_t)node * OUTC + j] = o[j] - lse;
}

extern "C" void kernel_launch(void* const* d_in, const int* in_sizes, int n_in,
                              void* d_out, int out_size, void* d_ws, size_t ws_size,
                              hipStream_t stream) {
  const float* x     = (const float*)d_in[0];
  const int*   ei    = (const int*)d_in[1];     // [2, NE]: row0 = src, row1 = dst
  const int*   src   = ei;
  const int*   dst   = ei + NE;
  // d_in[2] = edge_attr (unused by reference math)
  const float* l0_w1 = (const float*)d_in[3];
  const float* l0_b1 = (const float*)d_in[4];
  const float* l0_w2 = (const float*)d_in[5];
  const float* l0_b2 = (const float*)d_in[6];
  const float* l1_w1 = (const float*)d_in[7];
  const float* l1_b1 = (const float*)d_in[8];
  const float* l1_w2 = (const float*)d_in[9];
  const float* l1_b2 = (const float*)d_in[10];
  const float* l2_w1 = (const float*)d_in[11];
  const float* l2_b1 = (const float*)d_in[12];
  const float* l2_w2 = (const float*)d_in[13];
  const float* l2_b2 = (const float*)d_in[14];

  float* bufA = (float*)d_ws;                       // [NN, 64]
  float* bufB = bufA + (size_t)NN * HID;            // [NN, 64]
  float* outp = (float*)d_out;                      // [NN, 8]

  const int n4         = NN * HID / 4;
  const int copyBlocks = (n4 + 255) / 256;
  const int scatBlocks = (NE * 16) / 256;           // 25.6M threads exactly
  const int mlpBlocks  = (NN + 63) / 64;
  const int lastBlocks = (NN + 255) / 256;

  // ---- layer 0 ----
  copy_f4_kernel<<<copyBlocks, 256, 0, stream>>>((const float4*)x, (float4*)bufA, n4);
  scatter_add_kernel<<<scatBlocks, 256, 0, stream>>>(x, src, dst, bufA, NE);
  gin_mlp64_kernel<<<mlpBlocks, 128, 0, stream>>>(bufA, l0_w1, l0_b1, l0_w2, l0_b2,
                                                  bufB, NN, 1);
  // ---- layer 1 ----
  copy_f4_kernel<<<copyBlocks, 256, 0, stream>>>((const float4*)bufB, (float4*)bufA, n4);
  scatter_add_kernel<<<scatBlocks, 256, 0, stream>>>(bufB, src, dst, bufA, NE);
  gin_mlp64_kernel<<<mlpBlocks, 128, 0, stream>>>(bufA, l1_w1, l1_b1, l1_w2, l1_b2,
                                                  bufB, NN, 1);
  // ---- layer 2 + log_softmax ----
  copy_f4_kernel<<<copyBlocks, 256, 0, stream>>>((const float4*)bufB, (float4*)bufA, n4);
  scatter_add_kernel<<<scatBlocks, 256, 0, stream>>>(bufB, src, dst, bufA, NE);
  gin_last_kernel<<<lastBlocks, 256, 0, stream>>>(bufA, l2_w1, l2_b1, l2_w2, l2_b2,
                                                  outp, NN);
}